// OVDDetector_6098853560603
// MI455X (gfx1250) — compile-verified
//
#include <hip/hip_runtime.h>
#include <stdint.h>

// ---------------- problem constants (match reference) ----------------
constexpr int   kB          = 16;
constexpr int   kN          = 2048;
constexpr int   kCProto     = 512;
constexpr int   kNumClasses = 80;
constexpr int   kWords      = kN / 32;   // 64 mask words per row
constexpr float kImgW       = 560.0f;
constexpr float kImgH       = 560.0f;
constexpr float kMinBox     = 5.0f;
constexpr float kIouThr     = 0.1f;
constexpr float kConfThr    = 0.1f;
constexpr float kBoxConfThr = 0.1f;
constexpr float kMaxWh      = 4096.0f;

typedef __attribute__((ext_vector_type(2))) float v2f;
typedef __attribute__((ext_vector_type(8))) float v8f;

static __device__ __forceinline__ unsigned ballot32(bool p) {
  return (unsigned)__ballot((int)p);   // wave32: mask in low 32 bits
}

// ---------------------------------------------------------------------
// Kernel 1: per-proposal prep. One wave32 per proposal row.
// Coalesced read of 80 class scores, shuffle max+argmax (first occurrence),
// clip boxes, confidence gating.
// ---------------------------------------------------------------------
__global__ void ovd_prep(const float* __restrict__ proposals,
                         const float* __restrict__ preds,
                         const float* __restrict__ box_scores,
                         float* __restrict__ box0,
                         float* __restrict__ conf0,
                         int*   __restrict__ cls0) {
  int wid  = (blockIdx.x * blockDim.x + threadIdx.x) >> 5;  // proposal index
  int lane = threadIdx.x & 31;
  if (wid >= kB * kN) return;

  const float* row = preds + (size_t)wid * kCProto;
  float m = -1.0f; int mi = 0;
  for (int c = lane; c < kNumClasses; c += 32) {
    float v = row[c];
    if (v > m) { m = v; mi = c; }
  }
  for (int off = 16; off > 0; off >>= 1) {
    float om  = __shfl_xor(m,  off, 32);
    int   omi = __shfl_xor(mi, off, 32);
    if (om > m || (om == m && omi < mi)) { m = om; mi = omi; }
  }
  if (lane == 0) {
    const float* p = proposals + (size_t)wid * 4;
    float x1 = fminf(fmaxf(p[0], 0.0f), kImgW);
    float y1 = fminf(fmaxf(p[1], 0.0f), kImgH);
    float x2 = fminf(fmaxf(p[2], 0.0f), kImgW);
    float y2 = fminf(fmaxf(p[3], 0.0f), kImgH);
    float bs = box_scores[wid];
    bool valid = (bs > kBoxConfThr) && (x2 - x1 > kMinBox) && (y2 - y1 > kMinBox);
    float conf = bs * m;
    conf = (valid && conf > kConfThr) ? conf : 0.0f;
    box0[wid * 4 + 0] = x1;
    box0[wid * 4 + 1] = y1;
    box0[wid * 4 + 2] = x2;
    box0[wid * 4 + 3] = y2;
    conf0[wid] = conf;
    cls0[wid]  = mi;
  }
}

// ---------------------------------------------------------------------
// Kernel 2: per-image bitonic sort (descending by conf) of 2048 entries
// in LDS, then gather sorted box/conf/cls arrays.
// ---------------------------------------------------------------------
__global__ void ovd_sort(const float* __restrict__ conf0,
                         const float* __restrict__ box0,
                         const int*   __restrict__ cls0,
                         float* __restrict__ conf_s,
                         float* __restrict__ box_s,
                         int*   __restrict__ cls_s) {
  __shared__ float sc[kN];
  __shared__ int   si[kN];
  const int b = blockIdx.x;
  const int t = threadIdx.x;

  for (int i = t; i < kN; i += 1024) {
    sc[i] = conf0[b * kN + i];
    si[i] = i;
  }
  __syncthreads();

  for (int k = 2; k <= kN; k <<= 1) {
    for (int j = k >> 1; j > 0; j >>= 1) {
      for (int i = t; i < kN; i += 1024) {
        int ixj = i ^ j;
        if (ixj > i) {
          bool asc = (i & k) != 0;    // != 0 -> overall descending sort
          float a = sc[i], c = sc[ixj];
          bool doswap = asc ? (a > c) : (a < c);
          if (doswap) {
            sc[i] = c; sc[ixj] = a;
            int tmp = si[i]; si[i] = si[ixj]; si[ixj] = tmp;
          }
        }
      }
      __syncthreads();
    }
  }

  for (int n = t; n < kN; n += 1024) {
    int o = si[n];
    int g = b * kN + n, go = b * kN + o;
    conf_s[g] = sc[n];
    cls_s[g]  = cls0[go];
    box_s[g * 4 + 0] = box0[go * 4 + 0];
    box_s[g * 4 + 1] = box0[go * 4 + 1];
    box_s[g * 4 + 2] = box0[go * 4 + 2];
    box_s[g * 4 + 3] = box0[go * 4 + 3];
  }
}

// ---------------------------------------------------------------------
// Kernel 3: suppression bit-matrix. Block = 16 rows x 256 cols, 8 waves;
// each wave owns a 16x32 tile pair. Pairwise area sums (union base)
// computed on the matrix pipe with V_WMMA_F32_16X16X4_F32:
//   A row m = (area_i[m], 1, 0, 0), B col n = (1, area_j[n], 0, 0)
//   => D[m][n] = area_i[m] + area_j[n]  (exact f32, same as fadd)
// VALU does min/max intersection; wave32 ballot packs sup bits.
// ---------------------------------------------------------------------
__global__ void ovd_supmat(const float* __restrict__ box_s,
                           const int*   __restrict__ cls_s,
                           uint32_t*    __restrict__ supmat) {
  __shared__ float srx1[16], sry1[16], srx2[16], sry2[16], sra[16];
  __shared__ float scx1[256], scy1[256], scx2[256], scy2[256], sca[256];

  const int b  = blockIdx.z;
  const int rb = blockIdx.y * 16;    // row tile base
  const int cb = blockIdx.x * 256;   // col block base
  const int t  = threadIdx.x;

  const float* bs = box_s + (size_t)b * kN * 4;
  const int*   cs = cls_s + (size_t)b * kN;

  if (t < 16) {
    int i = rb + t;
    float o  = kMaxWh * (float)cs[i];
    float x1 = bs[i * 4 + 0] + o, y1 = bs[i * 4 + 1] + o;
    float x2 = bs[i * 4 + 2] + o, y2 = bs[i * 4 + 3] + o;
    srx1[t] = x1; sry1[t] = y1; srx2[t] = x2; sry2[t] = y2;
    sra[t] = fmaxf(x2 - x1, 0.0f) * fmaxf(y2 - y1, 0.0f);
  }
  {
    int j = cb + t;
    float o  = kMaxWh * (float)cs[j];
    float x1 = bs[j * 4 + 0] + o, y1 = bs[j * 4 + 1] + o;
    float x2 = bs[j * 4 + 2] + o, y2 = bs[j * 4 + 3] + o;
    scx1[t] = x1; scy1[t] = y1; scx2[t] = x2; scy2[t] = y2;
    sca[t] = fmaxf(x2 - x1, 0.0f) * fmaxf(y2 - y1, 0.0f);
  }
  __syncthreads();

  const int wave  = t >> 5;
  const int lane  = t & 31;
  const int ctile = wave * 32;            // this wave's 32-col window

  // Build WMMA operands per ISA 16x4 f32 A/B layout:
  // lanes 0-15 hold K=0,1; lanes 16-31 hold K=2,3 (set to zero).
  v2f A, B0, B1;
  A.x = 0.0f; A.y = 0.0f; B0 = A; B1 = A;
  if (lane < 16) {
    A.x  = sra[lane];            A.y  = 1.0f;
    B0.x = 1.0f;                 B0.y = sca[ctile + lane];
    B1.x = 1.0f;                 B1.y = sca[ctile + 16 + lane];
  }
  v8f Z = {};
  // (neg_a, A, neg_b, B, c_mod, C, reuse_a, reuse_b)
  v8f D0 = __builtin_amdgcn_wmma_f32_16x16x4_f32(false, A, false, B0, (short)0, Z, false, false);
  v8f D1 = __builtin_amdgcn_wmma_f32_16x16x4_f32(false, A, false, B1, (short)0, Z, false, false);

  // D layout: VGPR r -> lanes 0-15: (M=r, N=lane); lanes 16-31: (M=r+8, N=lane-16)
  const int nloc  = ctile + (lane & 15);
  const int mbase = (lane >> 4) * 8;
  const float cx1a = scx1[nloc],      cy1a = scy1[nloc];
  const float cx2a = scx2[nloc],      cy2a = scy2[nloc];
  const float cx1b = scx1[nloc + 16], cy1b = scy1[nloc + 16];
  const float cx2b = scx2[nloc + 16], cy2b = scy2[nloc + 16];
  const int j0 = cb + nloc;
  const int j1 = j0 + 16;
  const int wi = (cb >> 5) + wave;    // mask word index within row

  uint32_t* rowout = supmat + (size_t)b * kN * kWords;

#pragma unroll
  for (int r = 0; r < 8; ++r) {
    int m = mbase + r;
    int i = rb + m;
    float rx1 = srx1[m], ry1 = sry1[m], rx2 = srx2[m], ry2 = sry2[m];

    float ix1 = fmaxf(rx1, cx1a), iy1 = fmaxf(ry1, cy1a);
    float ix2 = fminf(rx2, cx2a), iy2 = fminf(ry2, cy2a);
    float inter0 = fmaxf(ix2 - ix1, 0.0f) * fmaxf(iy2 - iy1, 0.0f);
    float uni0   = D0[r] - inter0;
    bool s0 = (inter0 / (uni0 + 1e-7f) > kIouThr) && (j0 > i);

    ix1 = fmaxf(rx1, cx1b); iy1 = fmaxf(ry1, cy1b);
    ix2 = fminf(rx2, cx2b); iy2 = fminf(ry2, cy2b);
    float inter1 = fmaxf(ix2 - ix1, 0.0f) * fmaxf(iy2 - iy1, 0.0f);
    float uni1   = D1[r] - inter1;
    bool s1 = (inter1 / (uni1 + 1e-7f) > kIouThr) && (j1 > i);

    unsigned b0 = ballot32(s0);   // [15:0]=row r tile0, [31:16]=row r+8 tile0
    unsigned b1 = ballot32(s1);   // same for tile1 (cols +16)
    unsigned wlo = (b0 & 0xFFFFu) | (b1 << 16);            // row rb+r
    unsigned whi = (b0 >> 16) | (b1 & 0xFFFF0000u);        // row rb+r+8
    if (lane == 0) rowout[(size_t)(rb + r)     * kWords + wi] = wlo;
    if (lane == 1) rowout[(size_t)(rb + r + 8) * kWords + wi] = whi;
  }
}

// ---------------------------------------------------------------------
// Kernel 4: serial greedy scan, one wave32 per image. removed-mask in LDS
// (64 words, 2 per lane). 2048 iterations of: test bit i (uniform), if
// still kept OR its suppression row into removed.
// ---------------------------------------------------------------------
__global__ void ovd_scan(const float*    __restrict__ conf_s,
                         const uint32_t* __restrict__ supmat,
                         uint32_t*       __restrict__ remfin) {
  __shared__ uint32_t rem[kWords];
  const int b    = blockIdx.x;
  const int lane = threadIdx.x;

  for (int w = 0; w < kWords; ++w) {
    float c = conf_s[b * kN + w * 32 + lane];
    unsigned m = ballot32(c <= 0.0f);     // keep0 = conf > 0
    if (lane == 0) rem[w] = m;
  }
  __syncthreads();

  const uint32_t* rows = supmat + (size_t)b * kN * kWords;
  for (int i = 0; i < kN; ++i) {
    unsigned bit = (rem[i >> 5] >> (i & 31)) & 1u;   // uniform across wave
    if (!bit) {
      const uint32_t* row = rows + (size_t)i * kWords;
      rem[lane]      |= row[lane];
      rem[lane + 32] |= row[lane + 32];
    }
    __syncthreads();
  }
  remfin[b * kWords + lane]      = rem[lane];
  remfin[b * kWords + lane + 32] = rem[lane + 32];
}

// ---------------------------------------------------------------------
// Kernel 5: emit det rows [x1,y1,x2,y2,conf,cls] * keep
// ---------------------------------------------------------------------
__global__ void ovd_out(const float*    __restrict__ box_s,
                        const float*    __restrict__ conf_s,
                        const int*      __restrict__ cls_s,
                        const uint32_t* __restrict__ remfin,
                        float*          __restrict__ out) {
  int t = blockIdx.x * blockDim.x + threadIdx.x;
  if (t >= kB * kN) return;
  int b = t / kN, n = t % kN;
  unsigned bit = (remfin[b * kWords + (n >> 5)] >> (n & 31)) & 1u;
  float k = bit ? 0.0f : 1.0f;
  float* o = out + (size_t)t * 6;
  o[0] = box_s[t * 4 + 0] * k;
  o[1] = box_s[t * 4 + 1] * k;
  o[2] = box_s[t * 4 + 2] * k;
  o[3] = box_s[t * 4 + 3] * k;
  o[4] = conf_s[t] * k;
  o[5] = (float)cls_s[t] * k;
}

// ---------------------------------------------------------------------
extern "C" void kernel_launch(void* const* d_in, const int* in_sizes, int n_in,
                              void* d_out, int out_size, void* d_ws, size_t ws_size,
                              hipStream_t stream) {
  (void)in_sizes; (void)n_in; (void)out_size; (void)ws_size;
  const float* proposals  = (const float*)d_in[0];
  const float* preds      = (const float*)d_in[1];
  const float* box_scores = (const float*)d_in[2];
  float* out = (float*)d_out;

  char* w = (char*)d_ws;
  auto carve = [&](size_t bytes) -> char* {
    char* p = w;
    w += (bytes + 255) & ~(size_t)255;
    return p;
  };
  float*    box0   = (float*)   carve(sizeof(float) * kB * kN * 4);
  float*    conf0  = (float*)   carve(sizeof(float) * kB * kN);
  int*      cls0   = (int*)     carve(sizeof(int)   * kB * kN);
  float*    box_s  = (float*)   carve(sizeof(float) * kB * kN * 4);
  float*    conf_s = (float*)   carve(sizeof(float) * kB * kN);
  int*      cls_s  = (int*)     carve(sizeof(int)   * kB * kN);
  uint32_t* supmat = (uint32_t*)carve(sizeof(uint32_t) * (size_t)kB * kN * kWords); // 8 MB
  uint32_t* remfin = (uint32_t*)carve(sizeof(uint32_t) * kB * kWords);

  // 1 wave per proposal row: 32768 waves = 4096 blocks of 256
  ovd_prep<<<(kB * kN) / 8, 256, 0, stream>>>(proposals, preds, box_scores,
                                              box0, conf0, cls0);
  ovd_sort<<<kB, 1024, 0, stream>>>(conf0, box0, cls0, conf_s, box_s, cls_s);
  ovd_supmat<<<dim3(kN / 256, kN / 16, kB), 256, 0, stream>>>(box_s, cls_s, supmat);
  ovd_scan<<<kB, 32, 0, stream>>>(conf_s, supmat, remfin);
  ovd_out<<<(kB * kN) / 256, 256, 0, stream>>>(box_s, conf_s, cls_s, remfin, out);
}